// MISPool_68212670595726
// MI455X (gfx1250) — compile-verified
//
#include <hip/hip_runtime.h>

// ---------------------------------------------------------------------------
// MISPool for MI455X (gfx1250, wave32).
//   out = [ x_out : N*F floats | coarse : N*N floats ]
//   A symmetric  =>  coarse = diag(mis) * A^3 * diag(mis)
//   A^2 = A * A^T, A^3 = A^2 * A^T  -> one LDS-staged NT WMMA GEMM, called twice.
// ---------------------------------------------------------------------------

#define N_NODES 4096
#define FDIM    256
#define SENT    N_NODES
#define MIS_ITERS 64

typedef __attribute__((ext_vector_type(2))) float v2f;
typedef __attribute__((ext_vector_type(8))) float v8f;
typedef int v4i_vs __attribute__((vector_size(16)));   // matches builtin proto

// -------- CDNA5 async global->LDS path (guarded; falls back to ds_store) ----
#if defined(__has_builtin)
#if __has_builtin(__builtin_amdgcn_global_load_async_to_lds_b128) && \
    __has_builtin(__builtin_amdgcn_s_wait_asynccnt)
#define USE_ASYNC_LDS 1
#endif
#endif
#ifndef USE_ASYNC_LDS
#define USE_ASYNC_LDS 0
#endif

// ---------------------------- utility kernels ------------------------------

__global__ void zero_f32(float* __restrict__ p, long n) {
  long i = (long)blockIdx.x * blockDim.x + threadIdx.x;
  long stride = (long)gridDim.x * blockDim.x;
  for (; i < n; i += stride) p[i] = 0.0f;
}

__global__ void zero_i32(int* __restrict__ p, int n) {
  int i = blockIdx.x * blockDim.x + threadIdx.x;
  if (i < n) p[i] = 0;
}

// ------------------------- adjacency construction --------------------------

__global__ void build_adj(const int* __restrict__ row, const int* __restrict__ col,
                          const float* __restrict__ val, float* __restrict__ A, int E) {
  int e = blockIdx.x * blockDim.x + threadIdx.x;
  if (e < E) atomicAdd(&A[(long)row[e] * N_NODES + col[e]], val[e]);
}

// ------------------------------ MIS kernels --------------------------------

__global__ void mis_iter_init(int* __restrict__ nbr_min, int* __restrict__ rm, int n) {
  int i = blockIdx.x * blockDim.x + threadIdx.x;
  if (i < n) { nbr_min[i] = SENT; rm[i] = 0; }
}

__global__ void mis_scatter_min(const int* __restrict__ row, const int* __restrict__ col,
                                const int* __restrict__ removed, int* __restrict__ nbr_min, int E) {
  int e = blockIdx.x * blockDim.x + threadIdx.x;
  if (e < E) {
    int c = col[e];
    int r = removed[c] ? SENT : c;          // rank = node index
    atomicMin(&nbr_min[row[e]], r);
  }
}

__global__ void mis_select(const int* __restrict__ nbr_min, const int* __restrict__ removed,
                           int* __restrict__ sel, int n) {
  int i = blockIdx.x * blockDim.x + threadIdx.x;
  if (i < n) sel[i] = (!removed[i]) && (nbr_min[i] == i);
}

__global__ void mis_scatter_rm(const int* __restrict__ row, const int* __restrict__ col,
                               const int* __restrict__ sel, int* __restrict__ rm, int E) {
  int e = blockIdx.x * blockDim.x + threadIdx.x;
  if (e < E) {
    if (sel[col[e]]) atomicOr(&rm[row[e]], 1);
  }
}

__global__ void mis_update(const int* __restrict__ sel, const int* __restrict__ rm,
                           int* __restrict__ mis, int* __restrict__ removed, int n) {
  int i = blockIdx.x * blockDim.x + threadIdx.x;
  if (i < n) { mis[i] |= sel[i]; removed[i] |= rm[i]; }
}

// ------------------------------ pooling ------------------------------------

// one block per edge, one thread per feature
__global__ void pool_scatter(const int* __restrict__ row, const int* __restrict__ col,
                             const float* __restrict__ val, const float* __restrict__ x,
                             float* __restrict__ xout, int E) {
  int e = blockIdx.x;
  int f = threadIdx.x;
  if (e < E) {
    float v = x[(long)col[e] * FDIM + f] * val[e];
    atomicAdd(&xout[(long)row[e] * FDIM + f], v);
  }
}

__global__ void pool_mask(const int* __restrict__ mis, float* __restrict__ xout) {
  int i = blockIdx.x;
  int f = threadIdx.x;
  if (!mis[i]) xout[(long)i * FDIM + f] = 0.0f;
}

// --------------------------- WMMA f32 NT-GEMM ------------------------------
// Out = P * Q^T (N x N row-major).  Block = 256 threads (8 waves) computes a
// 128x128 tile; each wave a 32x64 sub-tile as 2x4 V_WMMA_F32_16X16X4_F32
// accumulators.  P/Q 128x32 panels staged in LDS (row stride 36 floats:
// 36*l mod 64 distinct for l=0..15 -> conflict-free b64 fragment reads;
// 144B row stride keeps b128 staging and b64 reads aligned).
//
// Fragment layouts (CDNA5 ISA 7.12.2, wave32):
//   A 16x4 f32 : lane l -> M=l, VGPR0=K0, VGPR1=K1 ; lane l+16 -> K2,K3
//   B 4x16 f32 : lane l -> N=l, VGPR0=K0, VGPR1=K1 ; lane l+16 -> K2,K3
//   C/D 16x16  : VGPR r, lanes 0-15 -> M=r ; lanes 16-31 -> M=8+r ; N=lane&15

#define BM 128
#define BN 128
#define BK 32
#define LDSW 36   // padded row stride (floats)

// copy 16 consecutive floats (64B) from global to LDS
__device__ __forceinline__ void copy16f_to_lds(const float* __restrict__ g, float* l) {
#if USE_ASYNC_LDS
  __attribute__((address_space(1))) v4i_vs* gp =
      (__attribute__((address_space(1))) v4i_vs*)(g);
  __attribute__((address_space(3))) v4i_vs* lp =
      (__attribute__((address_space(3))) v4i_vs*)(l);
  __builtin_amdgcn_global_load_async_to_lds_b128(gp,     lp,     0, 0);
  __builtin_amdgcn_global_load_async_to_lds_b128(gp + 1, lp + 1, 0, 0);
  __builtin_amdgcn_global_load_async_to_lds_b128(gp + 2, lp + 2, 0, 0);
  __builtin_amdgcn_global_load_async_to_lds_b128(gp + 3, lp + 3, 0, 0);
#else
  const float4* gs = (const float4*)g;
  float4* ls = (float4*)l;
  ls[0] = gs[0]; ls[1] = gs[1]; ls[2] = gs[2]; ls[3] = gs[3];
#endif
}

__device__ __forceinline__ void wait_async_lds() {
#if USE_ASYNC_LDS
  __builtin_amdgcn_s_wait_asynccnt(0);
#endif
}

__device__ __forceinline__ void store_tile(float* __restrict__ Out, v8f acc,
                                           int baseM, int baseN, int hf, int l,
                                           const int* __restrict__ mis, int maskFlag) {
#pragma unroll
  for (int r = 0; r < 8; ++r) {
    int om = baseM + hf * 8 + r;
    int on = baseN + l;
    float v = acc[r];
    if (maskFlag && !(mis[om] && mis[on])) v = 0.0f;
    Out[(long)om * N_NODES + on] = v;
  }
}

__global__ void __launch_bounds__(256)
gemm_nt_wmma(const float* __restrict__ P, const float* __restrict__ Q,
             float* __restrict__ Out, const int* __restrict__ mis, int maskFlag) {
  __shared__ float ldsP[BM * LDSW];
  __shared__ float ldsQ[BN * LDSW];

  const int tid  = threadIdx.x;
  const int lane = tid & 31;
  const int hf   = lane >> 4;       // selects K pair {0,1} vs {2,3}
  const int l    = lane & 15;
  const int w    = tid >> 5;        // wave 0..7
  const int wm   = (w & 3) * 32;    // wave M strip within block
  const int wn   = (w >> 2) * 64;   // wave N strip within block
  const int bM   = blockIdx.y * BM;
  const int bN   = blockIdx.x * BN;
  const int koff = 2 * hf;

  // staging: thread -> (panel row, 16-float half of the 32-wide K slab)
  const int sr = tid >> 1;          // 0..127
  const int sh = (tid & 1) << 4;    // 0 or 16
  const float* gP = P + (long)(bM + sr) * N_NODES + sh;
  const float* gQ = Q + (long)(bN + sr) * N_NODES + sh;
  float* lP = ldsP + sr * LDSW + sh;
  float* lQ = ldsQ + sr * LDSW + sh;

  v8f acc[2][4] = {};

  for (int kt = 0; kt < N_NODES; kt += BK) {
    __syncthreads();                          // previous tile's reads done
    copy16f_to_lds(gP + kt, lP);
    copy16f_to_lds(gQ + kt, lQ);
    wait_async_lds();
    __syncthreads();                          // panels visible to all waves

#pragma unroll
    for (int k = 0; k < BK; k += 4) {
      const float* pa = ldsP + (wm + l) * LDSW + k + koff;
      const float* pb = ldsQ + (wn + l) * LDSW + k + koff;
      v2f a0 = *(const v2f*)(pa);
      v2f a1 = *(const v2f*)(pa + 16 * LDSW);
      v2f b0 = *(const v2f*)(pb);
      v2f b1 = *(const v2f*)(pb + 16 * LDSW);
      v2f b2 = *(const v2f*)(pb + 32 * LDSW);
      v2f b3 = *(const v2f*)(pb + 48 * LDSW);
      acc[0][0] = __builtin_amdgcn_wmma_f32_16x16x4_f32(false, a0, false, b0, (short)0, acc[0][0], false, false);
      acc[0][1] = __builtin_amdgcn_wmma_f32_16x16x4_f32(false, a0, false, b1, (short)0, acc[0][1], false, false);
      acc[0][2] = __builtin_amdgcn_wmma_f32_16x16x4_f32(false, a0, false, b2, (short)0, acc[0][2], false, false);
      acc[0][3] = __builtin_amdgcn_wmma_f32_16x16x4_f32(false, a0, false, b3, (short)0, acc[0][3], false, false);
      acc[1][0] = __builtin_amdgcn_wmma_f32_16x16x4_f32(false, a1, false, b0, (short)0, acc[1][0], false, false);
      acc[1][1] = __builtin_amdgcn_wmma_f32_16x16x4_f32(false, a1, false, b1, (short)0, acc[1][1], false, false);
      acc[1][2] = __builtin_amdgcn_wmma_f32_16x16x4_f32(false, a1, false, b2, (short)0, acc[1][2], false, false);
      acc[1][3] = __builtin_amdgcn_wmma_f32_16x16x4_f32(false, a1, false, b3, (short)0, acc[1][3], false, false);
    }
  }

#pragma unroll
  for (int am = 0; am < 2; ++am)
#pragma unroll
    for (int bn = 0; bn < 4; ++bn)
      store_tile(Out, acc[am][bn], bM + wm + am * 16, bN + wn + bn * 16,
                 hf, l, mis, maskFlag);
}

// ------------------------------- launcher ----------------------------------

extern "C" void kernel_launch(void* const* d_in, const int* in_sizes, int n_in,
                              void* d_out, int out_size, void* d_ws, size_t ws_size,
                              hipStream_t stream) {
  const float* x   = (const float*)d_in[0];
  const int*   row = (const int*)d_in[1];
  const int*   col = (const int*)d_in[2];
  const float* val = (const float*)d_in[3];
  const int E = in_sizes[1];
  const int N = N_NODES;

  float* xout   = (float*)d_out;                       // N*F
  float* coarse = xout + (long)N * FDIM;               // N*N

  // workspace layout: A (64MB) | A2 (64MB) | int arrays
  float* A  = (float*)d_ws;
  float* A2 = A + (long)N * N;
  int* ib      = (int*)(A2 + (long)N * N);
  int* mis     = ib;
  int* removed = ib + N;       // contiguous with mis for joint zeroing
  int* nbr_min = ib + 2 * N;
  int* sel     = ib + 3 * N;
  int* rm      = ib + 4 * N;

  const int T = 256;

  // init (A accumulates, x_out accumulates, mis/removed persist across iters)
  zero_f32<<<2048, T, 0, stream>>>(A, (long)N * N);
  zero_f32<<<1024, T, 0, stream>>>(xout, (long)N * FDIM);
  zero_i32<<<(2 * N + T - 1) / T, T, 0, stream>>>(mis, 2 * N);

  // dense adjacency
  build_adj<<<(E + T - 1) / T, T, 0, stream>>>(row, col, val, A, E);

  // greedy parallel MIS, fixed iteration cap (converged iterations are no-ops)
  for (int it = 0; it < MIS_ITERS; ++it) {
    mis_iter_init<<<(N + T - 1) / T, T, 0, stream>>>(nbr_min, rm, N);
    mis_scatter_min<<<(E + T - 1) / T, T, 0, stream>>>(row, col, removed, nbr_min, E);
    mis_select<<<(N + T - 1) / T, T, 0, stream>>>(nbr_min, removed, sel, N);
    mis_scatter_rm<<<(E + T - 1) / T, T, 0, stream>>>(row, col, sel, rm, E);
    mis_update<<<(N + T - 1) / T, T, 0, stream>>>(sel, rm, mis, removed, N);
  }

  // pooled features, masked to MIS rows
  pool_scatter<<<E, FDIM, 0, stream>>>(row, col, val, x, xout, E);
  pool_mask<<<N, FDIM, 0, stream>>>(mis, xout);

  // coarse = diag(mis) * A^3 * diag(mis)  via two NT GEMMs (A symmetric)
  dim3 g(N / BN, N / BM);
  gemm_nt_wmma<<<g, 256, 0, stream>>>(A,  A, A2,     mis, 0);  // A^2 = A * A^T
  gemm_nt_wmma<<<g, 256, 0, stream>>>(A2, A, coarse, mis, 1);  // A^3, masked
}